// TopLoss_55190329754368
// MI455X (gfx1250) — compile-verified
//
#include <hip/hip_runtime.h>
#include <stdint.h>

// CDNA5 / gfx1250, wave32. Latency-optimized single workgroup:
//  - beta (16KB) staged into LDS via global_load_async_to_lds_b128 (ASYNCcnt),
//    overlapped with global_prefetch of the index arrays
//  - all random gathers hit LDS
//  - loss1 = sum(len1) - max(len1)  (no sort needed)
//  - intra-wave sum reduction through V_WMMA_F32_16X16X4_F32 (layout-agnostic
//    all-ones-B identity: sum(D) == 16 * sum(A)); shuffles for the max.

typedef __attribute__((ext_vector_type(2))) float v2f;  // A/B of 16x16x4 f32 WMMA
typedef __attribute__((ext_vector_type(8))) float v8f;  // C/D of 16x16 f32 WMMA

#define BLOCK  256
#define NWAVES (BLOCK / 32)
#define NEG_INF (-3.402823466e+38f)

__global__ __launch_bounds__(BLOCK)
void toploss_kernel(const float* __restrict__ beta, int nbeta,
                    const int* __restrict__ b0, const int* __restrict__ d0, int n0,
                    const int* __restrict__ b1, const int* __restrict__ d1, int n1,
                    float* __restrict__ out)
{
    __shared__ __align__(16) float sbeta[4096];  // 16 KB of the 320 KB/WGP LDS
    __shared__ float wsum[NWAVES];
    __shared__ float wmax[NWAVES];

    const int tid = threadIdx.x;

    if (nbeta == 4096) {   // block-uniform branch (kernel arg)
        // Async DMA-style stage: memory -> LDS directly, 16B per lane per issue,
        // no VGPR round-trip. Tracked by ASYNCcnt.
        const uint64_t gbase = (uint64_t)(uintptr_t)beta;
        #pragma unroll
        for (int i = tid * 4; i < 4096; i += BLOCK * 4) {
            uint32_t ldsa = (uint32_t)(uintptr_t)(&sbeta[i]);   // LDS aperture: low 32 bits
            uint64_t ga   = gbase + (uint64_t)i * 4u;
            asm volatile("global_load_async_to_lds_b128 %0, %1, off"
                         :: "v"(ldsa), "v"(ga) : "memory");
        }
        // Overlap: warm the index arrays into cache while ASYNCcnt drains.
        if (tid * 32 < n0) { __builtin_prefetch(&b0[tid * 32], 0, 0);
                             __builtin_prefetch(&d0[tid * 32], 0, 0); }
        if (tid * 32 < n1) { __builtin_prefetch(&b1[tid * 32], 0, 0);
                             __builtin_prefetch(&d1[tid * 32], 0, 0); }
        asm volatile("s_wait_asynccnt 0x0" ::: "memory");
    } else {
        for (int i = tid; i < 4096; i += BLOCK)
            sbeta[i] = (i < nbeta) ? beta[i] : 0.0f;
    }
    __syncthreads();

    float s  = 0.0f;     // combined partial sum: loss0 terms + len1 terms
    float mx = NEG_INF;  // partial max of len1

    for (int i = tid; i < n0; i += BLOCK)
        s += sbeta[b0[i]] - sbeta[d0[i]];

    for (int i = tid; i < n1; i += BLOCK) {
        float len = sbeta[b1[i]] - sbeta[d1[i]];
        s  += len;
        mx  = fmaxf(mx, len);
    }

    // ---- Intra-wave sum via WMMA (all 32 lanes active; EXEC all-ones here).
    // A: each lane contributes (s, 0) into its two A slots; B: all ones.
    // Identity: sum over all 256 D elements = 16 * sum over all A elements,
    // independent of the exact lane<->matrix mapping.
    v2f a; a[0] = s;    a[1] = 0.0f;
    v2f b; b[0] = 1.0f; b[1] = 1.0f;
    v8f c = {};
    v8f dm = __builtin_amdgcn_wmma_f32_16x16x4_f32(
        /*neg_a=*/false, a, /*neg_b=*/false, b,
        /*c_mod=*/(short)0, c, /*reuse_a=*/false, /*reuse_b=*/false);

    float acc = ((dm[0] + dm[1]) + (dm[2] + dm[3]))
              + ((dm[4] + dm[5]) + (dm[6] + dm[7]));

    // Cross-lane: sum(acc) over wave32 = 16 * wave_sum; max via shuffles.
    #pragma unroll
    for (int off = 16; off >= 1; off >>= 1) {
        acc += __shfl_xor(acc, off, 32);
        mx   = fmaxf(mx, __shfl_xor(mx, off, 32));
    }

    const int wave = tid >> 5;
    const int lane = tid & 31;
    if (lane == 0) {
        wsum[wave] = acc * (1.0f / 16.0f);
        wmax[wave] = mx;
    }
    __syncthreads();

    if (tid == 0) {
        float total = 0.0f;
        float m = NEG_INF;
        #pragma unroll
        for (int w = 0; w < NWAVES; ++w) {
            total += wsum[w];
            m      = fmaxf(m, wmax[w]);
        }
        if (n1 > 0) total -= m;   // skip-largest bar; n1==0 -> loss1 = 0
        out[0] = total;
    }
}

extern "C" void kernel_launch(void* const* d_in, const int* in_sizes, int n_in,
                              void* d_out, int out_size, void* d_ws, size_t ws_size,
                              hipStream_t stream) {
    const float* beta = (const float*)d_in[0];
    const int*   b0   = (const int*)d_in[1];
    const int*   d0   = (const int*)d_in[2];
    const int*   b1   = (const int*)d_in[3];
    const int*   d1   = (const int*)d_in[4];
    const int nbeta = in_sizes[0];
    const int n0    = in_sizes[1];
    const int n1    = in_sizes[3];

    toploss_kernel<<<1, BLOCK, 0, stream>>>(beta, nbeta, b0, d0, n0, b1, d1, n1,
                                            (float*)d_out);
}